// GuFilter_D_13116830122235
// MI455X (gfx1250) — compile-verified
//
#include <hip/hip_runtime.h>
#include <stdint.h>

// Depthwise 19x19 Gaussian blur, separable, fused through LDS.
// - Stage 1: GLOBAL_LOAD_ASYNC_TO_LDS_B32 (CDNA5 async DMA path, ASYNCcnt)
// - Stage 2/3: both 1-D passes as V_WMMA_F32_16X16X4_F32 against a constant
//   banded weight matrix held in registers (K=34 padded to 36, nine K=4 steps)
// - Output: nontemporal stores (write-once, keep L2 for halo reuse)
//
// Roofline: 512 MB min HBM traffic -> ~22us floor @ 23.3 TB/s; ~180 WMMA
// FLOP/output keeps the fp32 matrix pipe well under the memory floor.

typedef float v2f __attribute__((ext_vector_type(2)));
typedef float v8f __attribute__((ext_vector_type(8)));

#define KS        19
#define RAD       9
#define HW        512
#define TILE      64          // square output tile per workgroup
#define IN_ROWS   96          // 6 row-blocks of 16 (covers 64+18 = 82 halo rows)
#define IN_PITCH  84          // 82 halo cols + 2 overshoot for K=36; 20m mod 64 bank-clean
#define MID_ROWS  96
#define MID_PITCH 76          // 2*76 mod 64 = 24 (B loads), 8*76 mod 64 = 32 (C stores): clean
#define NRB_H     6
#define NRB_V     4
#define NCB       4

__global__ __launch_bounds__(256) void gauss_sep_wmma(
    const float* __restrict__ x, const float* __restrict__ sigma_p,
    float* __restrict__ out)
{
  __shared__ float sIn [IN_ROWS  * IN_PITCH ];
  __shared__ float sMid[MID_ROWS * MID_PITCH];

  const int tid  = threadIdx.x;
  const int lane = tid & 31;
  // Wave id is uniform across the wave: scalarize it so the task loops
  // compile to scalar branches (keeps EXEC all-ones for WMMA, no lane masks).
  const int wave = __builtin_amdgcn_readfirstlane(tid >> 5);
  const int half = lane >> 4;   // which 16-lane half of the wave
  const int n    = lane & 15;   // N (or M) index within the 16x16 tile

  const int tileC = blockIdx.x * TILE;
  const int tileR = blockIdx.y * TILE;
  const size_t plane = (size_t)blockIdx.z * (size_t)(HW * HW);
  const float* __restrict__ xp = x   + plane;
  float*       __restrict__ op = out + plane;

  // ---- Stage 1: haloed tile -> LDS via async DMA (zero-pad at borders).
  // In-bounds lanes issue GLOBAL_LOAD_ASYNC_TO_LDS_B32 (VDST = wave-relative
  // LDS offset = low 32 bits of the generic LDS pointer, ISA 10.2);
  // out-of-bounds lanes ds_store a zero (same thread owns the slot, and the
  // ds_store retires before that thread issues any async op).
  {
    const int r0 = tileR - RAD;
    const int c0 = tileC - RAD;
    for (int idx = tid; idx < IN_ROWS * IN_PITCH; idx += 256) {
      int r  = idx / IN_PITCH;
      int c  = idx - r * IN_PITCH;
      int gy = r0 + r;
      int gx = c0 + c;
      uint32_t ldsoff = (uint32_t)(uintptr_t)&sIn[idx];
      if (gy >= 0 && gy < HW && gx >= 0 && gx < HW) {
        const float* ga = &xp[gy * HW + gx];
        asm volatile("global_load_async_to_lds_b32 %0, %1, off"
                     :: "v"(ldsoff), "v"(ga) : "memory");
      } else {
        sIn[idx] = 0.0f;
      }
    }
  }

  // ---- Gaussian weights from sigma (overlaps with the async DMA above).
  // Separable product of normalized 1-D gaussians == normalized 2-D gaussian.
  const float sg     = sigma_p[0];
  const float inv2s2 = 1.0f / (2.0f * sg * sg);
  float s = 0.0f;
  #pragma unroll
  for (int k = 0; k < KS; ++k) {
    float d = (float)(k - RAD);
    s += __expf(-d * d * inv2s2);
  }
  const float invSum = 1.0f / s;

  // Banded weight operand, nine K=4 chunks (K padded 34 -> 36).
  // Horizontal pass uses it as B: B[t,x] = w[t-x], x = lane&15, t = 4q+j+2*half.
  // Vertical pass uses it as A:   A[m,k] = w[k-m], m = lane&15, k = 4q+j+2*half.
  // Per-lane values coincide -> one register array serves both passes.
  v2f band[9];
  #pragma unroll
  for (int q = 0; q < 9; ++q) {
    #pragma unroll
    for (int j = 0; j < 2; ++j) {
      int   t = 4 * q + j + 2 * half;
      int   i = t - n;
      float d = (float)(i - RAD);
      band[q][j] = (i >= 0 && i < KS) ? __expf(-d * d * inv2s2) * invSum : 0.0f;
    }
  }

  // Drain this wave's async DMA, then rendezvous with the other waves.
  asm volatile("s_wait_asynccnt 0x0" ::: "memory");
  __syncthreads();

  // ---- Stage 2: horizontal pass. 24 mid 16x16 tiles = 3 per wave.
  // A[m,k] = sIn[rb*16+m][cb*16+k]  (m = lane&15, k = 4q+j+2*half)
  #pragma unroll
  for (int it = 0; it < (NRB_H * NCB) / 8; ++it) {
    int task = wave + 8 * it;
    int rb = task >> 2, cb = task & 3;
    v8f acc = {};
    const float* ap = &sIn[(rb * 16 + n) * IN_PITCH + cb * 16 + 2 * half];
    #pragma unroll
    for (int q = 0; q < 9; ++q) {
      v2f a;
      a.x = ap[4 * q];
      a.y = ap[4 * q + 1];
      acc = __builtin_amdgcn_wmma_f32_16x16x4_f32(
          false, a, false, band[q], (short)0, acc, false, false);
    }
    // C layout: VGPR v -> row (v + 8*half), col n
    float* mp = &sMid[(rb * 16 + half * 8) * MID_PITCH + cb * 16 + n];
    #pragma unroll
    for (int v = 0; v < 8; ++v)
      mp[v * MID_PITCH] = acc[v];
  }
  __syncthreads();

  // ---- Stage 3: vertical pass. 16 output tiles = 2 per wave.
  // B[t,x] = sMid[rb*16 + t][cb*16 + x]  (t = 4q+j+2*half, x = lane&15)
  #pragma unroll
  for (int it = 0; it < (NRB_V * NCB) / 8; ++it) {
    int task = wave + 8 * it;
    int rb = task >> 2, cb = task & 3;
    v8f acc = {};
    const float* bp = &sMid[(rb * 16 + 2 * half) * MID_PITCH + cb * 16 + n];
    #pragma unroll
    for (int q = 0; q < 9; ++q) {
      v2f b;
      b.x = bp[(4 * q)     * MID_PITCH];
      b.y = bp[(4 * q + 1) * MID_PITCH];
      acc = __builtin_amdgcn_wmma_f32_16x16x4_f32(
          false, band[q], false, b, (short)0, acc, false, false);
    }
    // Write-once output: nontemporal stores preserve L2 for halo reuse.
    float* gp = &op[(size_t)(tileR + rb * 16 + half * 8) * HW + tileC + cb * 16 + n];
    #pragma unroll
    for (int v = 0; v < 8; ++v)
      __builtin_nontemporal_store(acc[v], &gp[(size_t)v * HW]);
  }
}

extern "C" void kernel_launch(void* const* d_in, const int* in_sizes, int n_in,
                              void* d_out, int out_size, void* d_ws, size_t ws_size,
                              hipStream_t stream) {
  const float* x     = (const float*)d_in[0];
  const float* sigma = (const float*)d_in[1];
  float* out         = (float*)d_out;
  dim3 grid(HW / TILE, HW / TILE, 16 * 16);  // (col tiles, row tiles, N*C planes)
  gauss_sep_wmma<<<grid, 256, 0, stream>>>(x, sigma, out);
}